// NoQmix_74560632258885
// MI455X (gfx1250) — compile-verified
//
#include <hip/hip_runtime.h>
#include <hip/hip_bf16.h>

#define N_AG   64
#define F_DIM  256
#define H_DIM  256
#define ALPHA  0.2f
#define FROW   132   // 128 floats per K-chunk row + 4 pad (row stride mod 64 banks = 4)
#define BROW   260   // 256 K per packed-B row + 4 pad (row stride mod 64 banks = 4)

typedef __attribute__((ext_vector_type(2)))  float v2f;
typedef __attribute__((ext_vector_type(8)))  float v8f;
typedef __attribute__((__vector_size__(4 * sizeof(int)))) int v4i;

typedef __attribute__((address_space(1))) v4i* gv4i_ptr;   // global b128 payload
typedef __attribute__((address_space(3))) v4i* lv4i_ptr;   // LDS b128 payload

#if __has_builtin(__builtin_amdgcn_global_load_async_to_lds_b128)
#define HAVE_ASYNC_LDS 1
#else
#define HAVE_ASYNC_LDS 0
#endif

__device__ __forceinline__ void wait_asynccnt0() {
#if __has_builtin(__builtin_amdgcn_s_wait_asynccnt)
  __builtin_amdgcn_s_wait_asynccnt(0);
#else
  asm volatile("s_wait_asynccnt 0x0" ::: "memory");
#endif
}

// ---------------------------------------------------------------------------
// Stage 0: w1 = W @ a[:H], w2 = W @ a[H:].  W is (F,H) row-major.
// ---------------------------------------------------------------------------
__global__ void wproj_kernel(const float* __restrict__ W,
                             const float* __restrict__ a,
                             float* __restrict__ wv) {
  const int f = threadIdx.x;
  const float* wr = W + f * H_DIM;
  float acc1 = 0.f, acc2 = 0.f;
#pragma unroll 8
  for (int h = 0; h < H_DIM; ++h) {
    float w = wr[h];
    acc1 = fmaf(w, a[h], acc1);
    acc2 = fmaf(w, a[H_DIM + h], acc2);
  }
  wv[f]         = acc1;
  wv[F_DIM + f] = acc2;
}

// ---------------------------------------------------------------------------
// Fused kernel: one workgroup (128 threads = 4 wave32) per batch element.
//   A) async-stage features[b] (64x256 f32) into LDS in two 32KB K-chunks
//   B) s1/s2 = feat . [w1 w2] via V_WMMA_F32_16X16X4_F32 (B pre-packed in LDS)
//   C) column softmax over i + q contribution, reduce, write out[b]
// ---------------------------------------------------------------------------
__global__ __launch_bounds__(128)
void gat_fused_kernel(const float* __restrict__ features,
                      const float* __restrict__ agent_qs,
                      const int*   __restrict__ adj,
                      const float* __restrict__ wv,
                      float* __restrict__ out) {
  __shared__ __align__(16) float lds_feat[N_AG * FROW];  // 33792 B
  __shared__ __align__(16) float lds_bp[16 * BROW];      // 16640 B packed B: [n][k]
  __shared__ float lds_s1[N_AG];
  __shared__ float lds_s2[N_AG];
  __shared__ float lds_red[N_AG];

  const int b    = blockIdx.x;
  const int tid  = threadIdx.x;
  const int lane = tid & 31;
  const int wave = tid >> 5;

  // ---- pre-pack WMMA B matrix: column 0 = w1, column 1 = w2, 2..15 = 0
  for (int i = tid; i < 16 * F_DIM; i += 128) {
    const int n = i >> 8;          // B column 0..15
    const int k = i & 255;         // K index
    float v = 0.f;
    if (n == 0)      v = wv[k];
    else if (n == 1) v = wv[F_DIM + k];
    lds_bp[n * BROW + k] = v;
  }

  const int   mbase = wave * 16;   // this wave's 16-row tile
  const int   mn    = lane & 15;   // A: row within tile; B/D: column index
  const int   khalf = lane >> 4;   // K interleave half
  const float* arow = lds_feat + (mbase + mn) * FROW;
  const float* brow = lds_bp + mn * BROW;
  const float4* src = (const float4*)(features + (size_t)b * (N_AG * F_DIM));

  v8f acc = {};

  for (int chunk = 0; chunk < 2; ++chunk) {
    // ---- stage A: K-chunk [chunk*128, chunk*128+128) -> LDS, b128 granules
#if HAVE_ASYNC_LDS
    for (int i = tid; i < N_AG * 32; i += 128) {
      const int r = i >> 5, c = i & 31;
      const float4* g = src + r * (F_DIM / 4) + chunk * 32 + c;
      float* l = &lds_feat[r * FROW + c * 4];
      __builtin_amdgcn_global_load_async_to_lds_b128(
          (gv4i_ptr)g, (lv4i_ptr)l, /*offset=*/0, /*cpol=*/0);
    }
    wait_asynccnt0();
#else
    for (int i = tid; i < N_AG * 32; i += 128) {
      const int r = i >> 5, c = i & 31;
      float4 v = src[r * (F_DIM / 4) + chunk * 32 + c];
      *(float4*)&lds_feat[r * FROW + c * 4] = v;
    }
#endif
    __syncthreads();

    const int kofs = chunk * 128;  // global K offset into packed B

    // ---- stage B: V_WMMA_F32_16X16X4_F32 accumulate
    // A 16x4 layout: lane holds row mn; VGPR0/1 = K = 2*khalf, 2*khalf+1
    // B 4x16 layout: lane holds col mn; VGPR0/1 = K = 2*khalf, 2*khalf+1
#pragma unroll 4
    for (int k0 = 0; k0 < 128; k0 += 4) {
      const int ka = k0 + 2 * khalf;
      v2f A  = *(const v2f*)&arow[ka];
      v2f Bv = *(const v2f*)&brow[kofs + ka];
      acc = __builtin_amdgcn_wmma_f32_16x16x4_f32(
                false, A, false, Bv, (short)0, acc, false, false);
    }
    __syncthreads();   // protect LDS tile before next chunk overwrites it
  }

  // D layout: lane holds column mn; VGPR v holds row mbase + v + 8*khalf
  if (mn < 2) {
    float* sd = (mn == 0) ? lds_s1 : lds_s2;
    const int mo = mbase + 8 * khalf;
#pragma unroll
    for (int v = 0; v < 8; ++v) sd[mo + v] = acc[v];
  }
  __syncthreads();

  // ---- stage C: per-column j softmax over i, then q contribution
  if (tid < N_AG) {
    const int   j   = tid;
    const float s2j = lds_s2[j];
    const float qj  = agent_qs[(size_t)b * N_AG + j];

    float mx = -3.402823466e38f;
    for (int i = 0; i < N_AG; ++i) {
      float e = lds_s1[i] + s2j;
      e = (e > 0.f) ? e : ALPHA * e;
      e = (adj[i * N_AG + j] > 0) ? e : -1.0e12f;
      mx = fmaxf(mx, e);
    }
    float z = 0.f;
    for (int i = 0; i < N_AG; ++i) {
      float e = lds_s1[i] + s2j;
      e = (e > 0.f) ? e : ALPHA * e;
      e = (adj[i * N_AG + j] > 0) ? e : -1.0e12f;
      z += __expf(e - mx);
    }
    const float rz = 1.f / z;
    float contrib = 0.f;
    for (int i = 0; i < N_AG; ++i) {
      float e = lds_s1[i] + s2j;
      e = (e > 0.f) ? e : ALPHA * e;
      e = (adj[i * N_AG + j] > 0) ? e : -1.0e12f;
      contrib = fmaf(__expf(e - mx) * rz, qj, contrib);
    }
    lds_red[j] = contrib;
  }
  __syncthreads();

  if (tid == 0) {
    float q = 0.f;
#pragma unroll
    for (int j = 0; j < N_AG; ++j) q += lds_red[j];
    out[b] = q;
  }
}

// ---------------------------------------------------------------------------
extern "C" void kernel_launch(void* const* d_in, const int* in_sizes, int n_in,
                              void* d_out, int out_size, void* d_ws, size_t ws_size,
                              hipStream_t stream) {
  const float* features = (const float*)d_in[0];
  const float* agent_qs = (const float*)d_in[1];
  const int*   adj      = (const int*)d_in[2];
  // d_in[3] = states: unused by the reference computation
  const float* W        = (const float*)d_in[4];
  const float* a        = (const float*)d_in[5];

  float* wv = (float*)d_ws;   // 512 floats: w1 | w2

  const int B = in_sizes[0] / (N_AG * F_DIM);

  wproj_kernel<<<1, H_DIM, 0, stream>>>(W, a, wv);
  gat_fused_kernel<<<B, 128, 0, stream>>>(features, agent_qs, adj, wv,
                                          (float*)d_out);
}